// LPSparseMAP_26276609916980
// MI455X (gfx1250) — compile-verified
//
#include <hip/hip_runtime.h>
#include <math.h>
#include <stdint.h>

#define BATCH 4096
#define DIM   8192
#define NSPL  31
#define NNOD  63
#define KC    64
#define XSTR  68   // LDS row stride (floats): 16B-aligned rows, conflict-free WMMA reads

typedef float v2f __attribute__((ext_vector_type(2)));
typedef float v8f __attribute__((ext_vector_type(8)));

// Async global->LDS 128-bit copy (CDNA5, ASYNCcnt-tracked). GV addressing mode.
__device__ __forceinline__ void async_ld_b128(unsigned lds_byte, const void* g) {
  asm volatile("global_load_async_to_lds_b128 %0, %1, off"
               :: "v"(lds_byte), "v"((unsigned long long)(uintptr_t)g));
}
__device__ __forceinline__ void wait_async0() {
  asm volatile("s_wait_asynccnt 0x0" ::: "memory");
}

// ------------------------------------------------------------------
// K1: XA = x @ A^T with V_WMMA_F32_16X16X4_F32 (fp32-exact), double-
//     buffered GLOBAL_LOAD_ASYNC_TO_LDS_B128 pipeline, fused tree-min
//     propagation -> q.  4 waves/block, 64 rows x 32 (padded) cols.
// ------------------------------------------------------------------
__global__ __launch_bounds__(128) void k1_gemm_q(const float* __restrict__ x,
                                                 const float* __restrict__ A,
                                                 float* __restrict__ q,    // [BATCH][NNOD]
                                                 float* __restrict__ qT)   // [NNOD][BATCH]
{
  __shared__ __align__(16) float xs [2][64][XSTR];
  __shared__ __align__(16) float wsA[2][32][XSTR];
  __shared__ float xa[64][33];

  const int tid  = threadIdx.x;
  const int lane = tid & 31;
  const int wv   = tid >> 5;                 // wave id 0..3
  const int rowBase = blockIdx.x * 64;

  const int mrow = lane & 15;                // M (or N) index within fragment
  const int koff = (lane >> 4) * 2;          // lanes 16..31 hold K+2,K+3

  // one-time zero of weight row 31 (both buffers); async loads never write it
  for (int v = tid; v < 2 * XSTR; v += 128) wsA[v / XSTR][31][v % XSTR] = 0.f;

  v8f acc0 = {};  // cols 0..15
  v8f acc1 = {};  // cols 16..31 (col 31 zero-padded)

  const unsigned xsBase = (unsigned)(uintptr_t)&xs[0][0][0];
  const unsigned waBase = (unsigned)(uintptr_t)&wsA[0][0][0];

  // issue one chunk of async copies into buffer `buf`
  auto issue = [&](int buf, int k0) {
    // x tile: 64 rows x KC floats -> 1024 b128 ops / block, 8 per thread
    for (int v = tid; v < 64 * (KC / 4); v += 128) {
      const int r = v >> 4;
      const int c = (v & 15) << 2;
      const unsigned lb = xsBase + (unsigned)(((buf * 64 + r) * XSTR + c) * 4);
      async_ld_b128(lb, x + (size_t)(rowBase + r) * DIM + k0 + c);
    }
    // A tile: 31 rows x KC floats (row 31 stays zero)
    for (int v = tid; v < 32 * (KC / 4); v += 128) {
      const int r = v >> 4;
      if (r < NSPL) {
        const int c = (v & 15) << 2;
        const unsigned lb = waBase + (unsigned)(((buf * 32 + r) * XSTR + c) * 4);
        async_ld_b128(lb, A + (size_t)r * DIM + k0 + c);
      }
    }
  };

  issue(0, 0);
  int cur = 0;
  for (int k0 = 0; k0 < DIM; k0 += KC) {
    wait_async0();        // this wave's async writes to xs/wsA[cur] have landed
    __syncthreads();      // ...and every other wave's too; also fences prior LDS reads
    if (k0 + KC < DIM) issue(cur ^ 1, k0 + KC);

    #pragma unroll
    for (int kk = 0; kk < KC; kk += 4) {
      v2f a, b0, b1;
      a.x  = xs[cur][wv * 16 + mrow][kk + koff];
      a.y  = xs[cur][wv * 16 + mrow][kk + koff + 1];
      b0.x = wsA[cur][mrow][kk + koff];
      b0.y = wsA[cur][mrow][kk + koff + 1];
      b1.x = wsA[cur][16 + mrow][kk + koff];
      b1.y = wsA[cur][16 + mrow][kk + koff + 1];
      acc0 = __builtin_amdgcn_wmma_f32_16x16x4_f32(false, a, false, b0, (short)0, acc0, false, false);
      acc1 = __builtin_amdgcn_wmma_f32_16x16x4_f32(false, a, false, b1, (short)0, acc1, false, false);
    }
    cur ^= 1;
  }

  // stage XA tile to LDS (C layout: VGPR j -> M=(lane>>4)*8+j, N=lane&15)
  __syncthreads();
  #pragma unroll
  for (int j = 0; j < 8; ++j) {
    const int r = (lane >> 4) * 8 + j;
    xa[wv * 16 + r][lane & 15]        = acc0[j];
    xa[wv * 16 + r][16 + (lane & 15)] = acc1[j];
  }
  __syncthreads();

  // q fixed point: q[t] = min(q[parent], (t odd ? +XA : -XA)[parent]); BFS order
  if (lane < 16) {
    const int r = wv * 16 + lane;
    float qv[NNOD];
    qv[0] = 1.0f;
    #pragma unroll
    for (int t = 1; t < NNOD; ++t) {
      const int p = (t - 1) >> 1;
      const float xv = xa[r][p];
      const float edge = (t & 1) ? xv : -xv;
      qv[t] = fminf(qv[p], edge);
    }
    const size_t grow = (size_t)(rowBase + r);
    #pragma unroll
    for (int t = 0; t < NNOD; ++t) {
      q [grow * NNOD + t]          = qv[t];
      qT[(size_t)t * BATCH + grow] = qv[t];
    }
  }
}

// ------------------------------------------------------------------
// K2: per-node column sort (descending, bitonic in LDS) + exclusive
//     prefix sums + initial per-node d via threshold bisection.
// ------------------------------------------------------------------
__global__ __launch_bounds__(256) void k2_sort_d0(const float* __restrict__ qT,
                                                  const float* __restrict__ eta,
                                                  float* __restrict__ sortedT,  // [NNOD][BATCH]
                                                  float* __restrict__ prefT,    // [NNOD][BATCH+1]
                                                  float* __restrict__ dInit)    // [NNOD]
{
  __shared__ float s[BATCH];
  __shared__ float pref[BATCH + 1];
  __shared__ float csum[257];
  const int n   = blockIdx.x;
  const int tid = threadIdx.x;

  for (int i = tid; i < BATCH; i += 256) s[i] = qT[(size_t)n * BATCH + i];
  __syncthreads();

  // bitonic sort, descending
  for (int k = 2; k <= BATCH; k <<= 1) {
    for (int j = k >> 1; j > 0; j >>= 1) {
      for (int i = tid; i < BATCH; i += 256) {
        const int ixj = i ^ j;
        if (ixj > i) {
          const float a = s[i], b = s[ixj];
          if (((i & k) == 0) ? (a < b) : (a > b)) { s[i] = b; s[ixj] = a; }
        }
      }
      __syncthreads();
    }
  }

  // exclusive prefix sum: pref[i] = sum of s[0..i-1]
  const int base = tid * (BATCH / 256);
  float loc = 0.f;
  for (int e = 0; e < BATCH / 256; ++e) loc += s[base + e];
  csum[tid] = loc;
  __syncthreads();
  if (tid == 0) {
    float run = 0.f;
    for (int i = 0; i < 256; ++i) { const float c = csum[i]; csum[i] = run; run += c; }
    pref[BATCH] = run;
  }
  __syncthreads();
  float run = csum[tid];
  for (int e = 0; e < BATCH / 256; ++e) { pref[base + e] = run; run += s[base + e]; }
  __syncthreads();

  for (int i = tid; i < BATCH; i += 256) {
    sortedT[(size_t)n * BATCH + i]     = s[i];
    prefT[(size_t)n * (BATCH + 1) + i] = pref[i];
  }
  if (tid == 0) prefT[(size_t)n * (BATCH + 1) + BATCH] = pref[BATCH];

  if (tid == 0) {
    float S = 0.f;
    for (int t = 0; t < NNOD; ++t) S += eta[t];
    const float d0 = eta[n];
    float lo = 0.f, hi = 1.f;
    for (int it = 0; it < 48; ++it) {
      const float mid = 0.5f * (lo + hi);
      const float T = fmaxf(mid, d0);
      int l = 0, h = BATCH;
      while (l < h) { const int m = (l + h) >> 1; if (s[m] >= T) l = m + 1; else h = m; }
      const float cand = (S + pref[l]) / (63.0f + (float)l);
      if (cand > mid) lo = mid; else hi = mid;
    }
    const float T = fmaxf(lo, d0);
    int l = 0, h = BATCH;
    while (l < h) { const int m = (l + h) >> 1; if (s[m] >= T) l = m + 1; else h = m; }
    dInit[n] = (l > 0) ? (S + pref[l]) / (63.0f + (float)l) : d0;
  }
}

// ------------------------------------------------------------------
// K3: sequential color-merge loop; each merged-color threshold found
//     by bisection with per-column binary searches on sorted columns.
// ------------------------------------------------------------------
__global__ __launch_bounds__(64) void k3_merge(const float* __restrict__ sortedT,
                                               const float* __restrict__ prefT,
                                               const float* __restrict__ eta,
                                               const float* __restrict__ dInit,
                                               float* __restrict__ dFin)
{
  __shared__ float d[NNOD];
  __shared__ float et[NNOD];
  __shared__ int   col[NNOD];
  __shared__ int   msk[NNOD];
  __shared__ float cArr[NNOD];
  __shared__ float sArr[NNOD];
  __shared__ float fctl[4];
  __shared__ int   ictl[4];
  const int tid = threadIdx.x;

  if (tid < NNOD) { d[tid] = dInit[tid]; et[tid] = eta[tid]; col[tid] = tid; }
  __syncthreads();
  float S = 0.f;
  for (int t = 0; t < NNOD; ++t) S += et[t];

  for (int iter = 0; iter < NNOD - 1; ++iter) {
    __syncthreads();
    if (tid == 0) {
      float best = -__builtin_inff(); int ts = -1;
      for (int t = 1; t < NNOD; ++t)
        if (d[t] > d[(t - 1) >> 1] && d[t] > best) { best = d[t]; ts = t; }
      ictl[0] = ts;
      if (ts >= 0) { ictl[1] = col[(ts - 1) >> 1]; ictl[2] = col[ts]; }
    }
    __syncthreads();
    const int ts = ictl[0];
    if (ts < 0) continue;                 // uniform: no further merges possible
    const int pc = ictl[1], tc = ictl[2];
    if (tid < NNOD) {
      if (col[tid] == tc) col[tid] = pc;
      msk[tid] = (col[tid] == pc) ? 1 : 0;
    }
    __syncthreads();
    if (tid == 0) {
      float se = 0.f, ce = 0.f;
      for (int t = 0; t < NNOD; ++t) if (msk[t]) { se += et[t]; ce += 1.f; }
      fctl[0] = se / ce;
    }
    __syncthreads();
    const float d0 = fctl[0];
    float lo = 0.f, hi = 1.f, dnew = d0;
    for (int it = 0; it <= 48; ++it) {
      const float mid = (it < 48) ? 0.5f * (lo + hi) : lo;
      const float T = fmaxf(mid, d0);
      if (tid < NNOD) {
        float cc = 0.f, ss = 0.f;
        if (msk[tid]) {
          const float* srow = sortedT + (size_t)tid * BATCH;
          int l = 0, h = BATCH;
          while (l < h) { const int m = (l + h) >> 1; if (srow[m] >= T) l = m + 1; else h = m; }
          cc = (float)l;
          ss = prefT[(size_t)tid * (BATCH + 1) + l];
        }
        cArr[tid] = cc; sArr[tid] = ss;
      }
      __syncthreads();
      if (tid == 0) {
        float ct = 0.f, st = 0.f;
        for (int t = 0; t < NNOD; ++t) { ct += cArr[t]; st += sArr[t]; }
        fctl[1] = (S + st) / (63.0f + ct);
        fctl[2] = ct;
      }
      __syncthreads();
      const float cand = fctl[1];
      if (it < 48) { if (cand > mid) lo = mid; else hi = mid; }
      else         { dnew = (fctl[2] > 0.f) ? cand : d0; }
      __syncthreads();
    }
    if (tid < NNOD && msk[tid]) d[tid] = dnew;
  }
  __syncthreads();
  if (tid < NNOD) dFin[tid] = fminf(fmaxf(d[tid], 0.f), 1.f);
}

// ------------------------------------------------------------------
// K4: out = min(clip(q,0,1), d[node])
// ------------------------------------------------------------------
__global__ __launch_bounds__(256) void k4_out(const float* __restrict__ q,
                                              const float* __restrict__ dFin,
                                              float* __restrict__ out)
{
  const int idx = blockIdx.x * 256 + threadIdx.x;
  if (idx < BATCH * NNOD) {
    const int n = idx % NNOD;
    const float z = fminf(fmaxf(q[idx], 0.f), 1.f);
    out[idx] = fminf(z, dFin[n]);
  }
}

extern "C" void kernel_launch(void* const* d_in, const int* in_sizes, int n_in,
                              void* d_out, int out_size, void* d_ws, size_t ws_size,
                              hipStream_t stream) {
  (void)in_sizes; (void)n_in; (void)out_size; (void)ws_size;
  const float* x   = (const float*)d_in[0];
  const float* A   = (const float*)d_in[1];
  const float* eta = (const float*)d_in[2];
  float* out = (float*)d_out;

  float* w      = (float*)d_ws;
  float* q      = w;                               // BATCH*NNOD
  float* qT     = q      + (size_t)BATCH * NNOD;   // NNOD*BATCH
  float* sorted = qT     + (size_t)NNOD * BATCH;   // NNOD*BATCH
  float* prefT  = sorted + (size_t)NNOD * BATCH;   // NNOD*(BATCH+1)
  float* dInit  = prefT  + (size_t)NNOD * (BATCH + 1);
  float* dFin   = dInit  + NNOD;

  k1_gemm_q<<<BATCH / 64, 128, 0, stream>>>(x, A, q, qT);
  k2_sort_d0<<<NNOD, 256, 0, stream>>>(qT, eta, sorted, prefT, dInit);
  k3_merge<<<1, 64, 0, stream>>>(sorted, prefT, eta, dInit, dFin);
  k4_out<<<(BATCH * NNOD + 255) / 256, 256, 0, stream>>>(q, dFin, out);
}